// LNCC_45973329936798
// MI455X (gfx1250) — compile-verified
//
#include <hip/hip_runtime.h>
#include <hip/hip_bf16.h>
#include <math.h>

typedef __attribute__((ext_vector_type(2))) float v2f;
typedef __attribute__((ext_vector_type(8))) float v8f;

#define IMG   512
#define NIMG  32            // 16 batch * 2 channels
#define WS    9
#define PADR  4
#define TILE  32            // block output tile (32x32)
#define STG   40            // TILE + 2*PADR
#define EPSC  1e-8f
#define TOTAL 8388608.0f    // 16*2*512*512

// CDNA5 async global->LDS path (ASYNCcnt), with graceful fallback so the
// host pass / older toolchains still parse & compile.
#if __has_builtin(__builtin_amdgcn_global_load_async_to_lds_b32)
#define HAVE_ASYNC_LDS 1
typedef __attribute__((address_space(1))) float GFloat;
typedef __attribute__((address_space(3))) float LFloat;
typedef __attribute__((address_space(1))) int   GInt;
typedef __attribute__((address_space(3))) int   LInt;
#endif

__global__ void lncc_init(float* acc) { acc[0] = 0.0f; }

__global__ void lncc_final(const float* acc, float* out) {
    out[0] = -acc[0] / TOTAL;
}

__global__ __launch_bounds__(128)
void lncc_main(const float* __restrict__ fixed_, const float* __restrict__ moving_,
               float* __restrict__ acc) {
    __shared__ float sF[STG][STG + 1];
    __shared__ float sM[STG][STG + 1];
    __shared__ float sH[5][STG][TILE + 1];
    __shared__ float sRed;

    const int t    = threadIdx.x;      // 0..127
    const int wave = t >> 5;           // 0..3
    const int lane = t & 31;

    const int img = blockIdx.z;                 // 0..31
    const int by0 = blockIdx.y * TILE;          // output row base
    const int bx0 = blockIdx.x * TILE;          // output col base
    const size_t base = (size_t)img * IMG * IMG;

    if (t == 0) sRed = 0.0f;

    // Warm the L2 path for this block's tile (global_prefetch_b8).
    {
        int pr = by0 + (t % STG) - PADR;
        if (pr < 0) pr = 0; if (pr >= IMG) pr = IMG - 1;
        int pc = bx0 >= PADR ? bx0 - PADR : 0;
        __builtin_prefetch(fixed_  + base + (size_t)pr * IMG + pc, 0, 3);
        __builtin_prefetch(moving_ + base + (size_t)pr * IMG + pc, 0, 3);
    }

    // ---- Stage haloed 40x40 tiles of fixed/moving with zero SAME padding ----
    for (int idx = t; idx < STG * STG; idx += 128) {
        const int r = idx / STG, c = idx % STG;
        const int gr = by0 + r - PADR;
        const int gc = bx0 + c - PADR;
        const bool in = (gr >= 0) & (gr < IMG) & (gc >= 0) & (gc < IMG);
        const size_t off = base + (size_t)gr * IMG + gc;
#ifdef HAVE_ASYNC_LDS
        if (in) {
            // Per-lane async DMA, bypasses VGPRs, tracked by ASYNCcnt.
            __builtin_amdgcn_global_load_async_to_lds_b32(
                (GInt*)(GFloat*)(fixed_ + off),  (LInt*)(LFloat*)&sF[r][c], 0, 0);
            __builtin_amdgcn_global_load_async_to_lds_b32(
                (GInt*)(GFloat*)(moving_ + off), (LInt*)(LFloat*)&sM[r][c], 0, 0);
        } else {
            sF[r][c] = 0.0f;
            sM[r][c] = 0.0f;
        }
#else
        sF[r][c] = in ? fixed_[off]  : 0.0f;
        sM[r][c] = in ? moving_[off] : 0.0f;
#endif
    }
#ifdef HAVE_ASYNC_LDS
#if __has_builtin(__builtin_amdgcn_s_wait_asynccnt)
    __builtin_amdgcn_s_wait_asynccnt(0);
#else
    asm volatile("s_wait_asynccnt 0" ::: "memory");
#endif
#endif
    __syncthreads();

    // ---- Horizontal 9-tap sums of 5 product images: sliding-window update ----
    // 80 active threads; thread = (row, half). Each streams 16 outputs with a
    // running window (+new - old), ~3x fewer VALU/DS ops than naive 9-tap.
    if (t < 2 * STG) {
        const int r     = t >> 1;
        const int cbase = (t & 1) * 16;
        float a0 = 0.f, a1 = 0.f, a2 = 0.f, a3 = 0.f, a4 = 0.f;
#pragma unroll
        for (int i = 0; i < WS; ++i) {
            const float f = sF[r][cbase + i];
            const float m = sM[r][cbase + i];
            a0 += f; a1 += m; a2 += f * f; a3 += m * m; a4 += f * m;
        }
        sH[0][r][cbase] = a0; sH[1][r][cbase] = a1;
        sH[2][r][cbase] = a2; sH[3][r][cbase] = a3; sH[4][r][cbase] = a4;
#pragma unroll
        for (int c = 1; c < 16; ++c) {
            const float fn = sF[r][cbase + c + 8], mn_ = sM[r][cbase + c + 8];
            const float fo = sF[r][cbase + c - 1], mo  = sM[r][cbase + c - 1];
            a0 += fn - fo;
            a1 += mn_ - mo;
            a2 += fn * fn - fo * fo;
            a3 += mn_ * mn_ - mo * mo;
            a4 += fn * mn_ - fo * mo;
            sH[0][r][cbase + c] = a0; sH[1][r][cbase + c] = a1;
            sH[2][r][cbase + c] = a2; sH[3][r][cbase + c] = a3;
            sH[4][r][cbase + c] = a4;
        }
    }
    __syncthreads();

    // ---- Vertical 9-tap sum as banded matmul: D(16x16) = A(16x24) * H(24x16) ----
    // 6 chunks of V_WMMA_F32_16X16X4_F32 per image, 5 images.
    const int tr = (wave >> 1) & 1;    // tile row within block (0..1)
    const int tc = wave & 1;           // tile col within block (0..1)
    const int mn = lane & 15;          // A: M index; B: N index
    const int hi = lane >> 4;          // 0 -> K{0,1}; 1 -> K{2,3}

    v8f d0 = {}, d1 = {}, d2 = {}, d3 = {}, d4 = {};
#pragma unroll
    for (int kc = 0; kc < 6; ++kc) {
        const int k0 = kc * 4 + hi * 2;          // this lane's first K of the chunk
        // Band matrix: A[m][k] = 1 iff m <= k <= m+8  (24-row window, halo 4)
        v2f a;
        a.x = (k0     >= mn && k0     <= mn + 8) ? 1.0f : 0.0f;
        a.y = (k0 + 1 >= mn && k0 + 1 <= mn + 8) ? 1.0f : 0.0f;

        const int hr = tr * 16 + k0;             // H row for this lane's K
        const int hc = tc * 16 + mn;             // H col = N
        v2f b0, b1, b2, b3, b4;
        b0.x = sH[0][hr][hc]; b0.y = sH[0][hr + 1][hc];
        b1.x = sH[1][hr][hc]; b1.y = sH[1][hr + 1][hc];
        b2.x = sH[2][hr][hc]; b2.y = sH[2][hr + 1][hc];
        b3.x = sH[3][hr][hc]; b3.y = sH[3][hr + 1][hc];
        b4.x = sH[4][hr][hc]; b4.y = sH[4][hr + 1][hc];

        d0 = __builtin_amdgcn_wmma_f32_16x16x4_f32(false, a, false, b0, (short)0, d0, false, false);
        d1 = __builtin_amdgcn_wmma_f32_16x16x4_f32(false, a, false, b1, (short)0, d1, false, false);
        d2 = __builtin_amdgcn_wmma_f32_16x16x4_f32(false, a, false, b2, (short)0, d2, false, false);
        d3 = __builtin_amdgcn_wmma_f32_16x16x4_f32(false, a, false, b3, (short)0, d3, false, false);
        d4 = __builtin_amdgcn_wmma_f32_16x16x4_f32(false, a, false, b4, (short)0, d4, false, false);
    }

    // ---- Pointwise LNCC + local sum over this lane's 8 accumulator rows ----
    const float inv = 1.0f / 81.0f;
    float lsum = 0.0f;
#pragma unroll
    for (int j = 0; j < 8; ++j) {
        const float fmean = d0[j] * inv;
        const float mmean = d1[j] * inv;
        const float fsq   = d2[j] * inv;
        const float msq   = d3[j] * inv;
        const float fmv   = d4[j] * inv;
        const float fvar  = fsq - fmean * fmean;
        const float mvar  = msq - mmean * mmean;
        const float covar = fmv - fmean * mmean;
        const float denom = sqrtf(fvar + EPSC) * sqrtf(mvar + EPSC) + EPSC;
        lsum += covar / denom;
    }

    // ---- wave32 shuffle reduction, then LDS + global float atomics ----
#pragma unroll
    for (int off = 16; off > 0; off >>= 1)
        lsum += __shfl_xor(lsum, off, 32);
    if (lane == 0) atomicAdd(&sRed, lsum);
    __syncthreads();
    if (t == 0) atomicAdd(acc, sRed);
}

extern "C" void kernel_launch(void* const* d_in, const int* in_sizes, int n_in,
                              void* d_out, int out_size, void* d_ws, size_t ws_size,
                              hipStream_t stream) {
    const float* fixed_  = (const float*)d_in[0];
    const float* moving_ = (const float*)d_in[1];
    float* out = (float*)d_out;
    float* acc = (float*)d_ws;

    lncc_init<<<1, 1, 0, stream>>>(acc);
    dim3 grid(IMG / TILE, IMG / TILE, NIMG);   // 16 x 16 x 32
    lncc_main<<<grid, 128, 0, stream>>>(fixed_, moving_, acc);
    lncc_final<<<1, 1, 0, stream>>>(acc, out);
}